// OuterProductNetwork_3126736191879
// MI455X (gfx1250) — compile-verified
//
#include <hip/hip_runtime.h>

// OuterProductNetwork: out[b,n] = sum_{o,i} x[b,col_n,o] * kernel[o,n,i] * x[b,row_n,i]
// BATCH=8192, NUM_FIELDS=32, EMBED=64, NUM_IX=496.
// Compute-bound (~33 GFLOP fp32 vs ~88MB traffic) -> use V_WMMA_F32_16X16X4_F32.

typedef __attribute__((ext_vector_type(2))) float v2f;
typedef __attribute__((ext_vector_type(4))) float v4f;
typedef __attribute__((ext_vector_type(8))) float v8f;

#define BATCH      8192
#define NUM_FIELDS 32
#define EMBED      64
#define NUM_IX     496
#define LDS_STRIDE 68      // 64 + 4 pad: 16B-aligned rows, conflict-free b64 reads

__global__ __launch_bounds__(256) void opn_wmma_kernel(const float* __restrict__ x,
                                                       const float* __restrict__ kern,
                                                       float* __restrict__ out)
{
    __shared__ float Ksh[EMBED * LDS_STRIDE];

    const int n = blockIdx.x;                 // pair index 0..495 (uniform)

    // Decode triu pair (r,c), k=1, row-major -- scalar/uniform, EXEC untouched.
    int r = 0, nn = n;
    while (nn >= (NUM_FIELDS - 1 - r)) { nn -= (NUM_FIELDS - 1 - r); ++r; }
    const int c = r + 1 + nn;

    const int tid    = threadIdx.x;           // 0..255
    const int lane   = tid & 31;
    const int wave   = tid >> 5;              // 0..7
    const int laneLo = lane & 15;
    const int laneHi = lane >> 4;             // 0 or 1

    // ---- stage kernel[:, n, :] (64x64 f32 = 16KB) into LDS ----
    // kern element (o,n,i) at ((o*NUM_IX)+n)*EMBED + i ; row of 64 contiguous floats.
    {
        const int o  = tid >> 2;              // 0..63
        const int q4 = tid & 3;               // 16-float quarter of the row
        const float* src = kern + ((size_t)o * NUM_IX + n) * EMBED + q4 * 16;
        float*       dst = &Ksh[o * LDS_STRIDE + q4 * 16];
        #pragma unroll
        for (int j = 0; j < 4; ++j)
            *(v4f*)(dst + j * 4) = *(const v4f*)(src + j * 4);
    }
    __syncthreads();

    // ---- hoist B fragments (K_n^T: K=i, N=o) into registers, reused 64x ----
    // ISA layout (4x16 f32 B): lanes 0-15 hold K={0,1} (VGPR0,1) at N=lane;
    // lanes 16-31 hold K={2,3} at N=lane-16.  -> float2 at Ksh[o][i], i even.
    v2f Bf[16][4];
    #pragma unroll
    for (int k = 0; k < 16; ++k)
        #pragma unroll
        for (int t = 0; t < 4; ++t) {
            const int o = t * 16 + laneLo;
            const int i = k * 4 + 2 * laneHi;
            Bf[k][t] = *(const v2f*)&Ksh[o * LDS_STRIDE + i];
        }

    const size_t rowStride = (size_t)NUM_FIELDS * EMBED;  // 2048 floats per batch row

    // Each wave owns batch tiles btile = wave + 8*it, it = 0..63 (512 tiles total).
    #pragma unroll 1
    for (int it = 0; it < 64; ++it) {
        const int b0 = (wave + it * 8) * 16;

        // ---- A fragments: P tile (M=batch 16, K=i 4) per ISA 16x4 f32 layout ----
        // lane holds A[M=laneLo, K = 4k + 2*laneHi + {0,1}] = x[b0+laneLo, r, ...]
        const float* pA = x + (size_t)(b0 + laneLo) * rowStride + (size_t)r * EMBED + 2 * laneHi;
        v2f Af[16];
        #pragma unroll
        for (int k = 0; k < 16; ++k)
            Af[k] = *(const v2f*)(pA + k * 4);

        // ---- Q in C/D layout: element (VGPR rr, lane) = q[b0 + rr + 8*laneHi, o=laneLo+16t]
        const float* pQ = x + (size_t)(b0 + 8 * laneHi) * rowStride + (size_t)c * EMBED + laneLo;
        float Qf[4][8];
        #pragma unroll
        for (int t = 0; t < 4; ++t)
            #pragma unroll
            for (int rr = 0; rr < 8; ++rr)
                Qf[t][rr] = pQ[(size_t)rr * rowStride + t * 16];

        // ---- T[b, o] accumulation: 16 K-steps x 4 o-tiles of v_wmma_f32_16x16x4_f32
        v8f D[4] = {v8f{}, v8f{}, v8f{}, v8f{}};
        #pragma unroll
        for (int k = 0; k < 16; ++k) {
            #pragma unroll
            for (int t = 0; t < 4; ++t) {
                D[t] = __builtin_amdgcn_wmma_f32_16x16x4_f32(
                    /*neg_a=*/false, Af[k],
                    /*neg_b=*/false, Bf[k][t],
                    /*c_mod=*/(short)0, D[t],
                    /*reuse_a=*/false, /*reuse_b=*/false);
            }
        }

        // ---- out[b] = sum_o T[b,o]*q[b,o] : lane-local partials, then 16-lane butterfly
        float partial[8];
        #pragma unroll
        for (int rr = 0; rr < 8; ++rr) {
            float s = D[0][rr] * Qf[0][rr] + D[1][rr] * Qf[1][rr]
                    + D[2][rr] * Qf[2][rr] + D[3][rr] * Qf[3][rr];
            #pragma unroll
            for (int m = 1; m <= 8; m <<= 1)
                s += __shfl_xor(s, m, 32);
            partial[rr] = s;
        }

        if (laneLo == 0) {
            #pragma unroll
            for (int rr = 0; rr < 8; ++rr)
                out[(size_t)(b0 + 8 * laneHi + rr) * NUM_IX + n] = partial[rr];
        }
    }
}

extern "C" void kernel_launch(void* const* d_in, const int* in_sizes, int n_in,
                              void* d_out, int out_size, void* d_ws, size_t ws_size,
                              hipStream_t stream) {
    const float* x    = (const float*)d_in[0];   // (8192, 32, 64) f32
    const float* kern = (const float*)d_in[1];   // (64, 496, 64) f32
    float* out        = (float*)d_out;           // (8192, 496) f32

    dim3 grid(NUM_IX);    // one workgroup per field pair
    dim3 block(256);      // 8 wave32 waves
    hipLaunchKernelGGL(opn_wmma_kernel, grid, block, 0, stream, x, kern, out);
}